// DiffGuidedFilter_35948876267930
// MI455X (gfx1250) — compile-verified
//
#include <hip/hip_runtime.h>
#include <stdint.h>

#ifndef __has_builtin
#define __has_builtin(x) 0
#endif

#define GF_H   1024
#define GF_W   1024
#define TILE   32            // 32x32 outputs per block
#define IN_T   36            // input tile: TILE + 2*2 (two stencil levels, r=1 each)
#define AB_T   34            // a/b tile:   TILE + 2
#define NTH    256           // 8 wave32 waves
#define EPS_F  1e-8f

// Async global -> LDS copy of one 32-bit word per active lane (gfx1250 ASYNCcnt path).
__device__ __forceinline__ void gf_async_g2l(const float* gsrc, float* lds) {
#if defined(__HIP_DEVICE_COMPILE__)
  typedef __attribute__((address_space(1))) int GlobalI32;  // builtin is typed over int*
  typedef __attribute__((address_space(3))) int LdsI32;
#if __has_builtin(__builtin_amdgcn_global_load_async_to_lds_b32)
  __builtin_amdgcn_global_load_async_to_lds_b32(
      (GlobalI32*)(uintptr_t)gsrc,
      (LdsI32*)(uint32_t)(uintptr_t)lds,   // generic LDS addr low 32 bits == LDS offset
      0, 0);
#else
  asm volatile("global_load_async_to_lds_b32 %0, %1, off"
               :: "v"((uint32_t)(uintptr_t)lds), "v"((uint64_t)(uintptr_t)gsrc)
               : "memory");
#endif
#else
  (void)gsrc; (void)lds;   // host parse stub; never executed
#endif
}

__device__ __forceinline__ void gf_wait_async() {
#if defined(__HIP_DEVICE_COMPILE__)
#if __has_builtin(__builtin_amdgcn_s_wait_asynccnt)
  __builtin_amdgcn_s_wait_asynccnt(0);
#else
  asm volatile("s_wait_asynccnt 0x0" ::: "memory");
#endif
#endif
}

// count_include_pad=False reciprocal of valid-count; cnt ∈ {4,6,9}
__device__ __forceinline__ float gf_inv_cnt(int y, int x) {
  float cy = (y == 0 || y == GF_H - 1) ? 0.5f : (1.0f / 3.0f);
  float cx = (x == 0 || x == GF_W - 1) ? 0.5f : (1.0f / 3.0f);
  return cy * cx;
}

__global__ __launch_bounds__(NTH)
void guided_filter_fused(const float* __restrict__ g,
                         const float* __restrict__ s,
                         float* __restrict__ out) {
  __shared__ float sg [IN_T * IN_T];   // guidance tile (zero-padded at image edge)
  __shared__ float ssr[IN_T * IN_T];   // src tile
  __shared__ float rg [IN_T * AB_T];   // horizontal 3-tap sums
  __shared__ float rs [IN_T * AB_T];
  __shared__ float rgg[IN_T * AB_T];
  __shared__ float rgs[IN_T * AB_T];
  __shared__ float sa [AB_T * AB_T];   // a, b (zero outside image)
  __shared__ float sb [AB_T * AB_T];
  __shared__ float ra [AB_T * TILE];   // horizontal 3-tap sums of a, b
  __shared__ float rb [AB_T * TILE];

  const int tid = threadIdx.x;
  const int X0  = (int)blockIdx.x * TILE - 2;       // input-tile origin in image coords
  const int Y0  = (int)blockIdx.y * TILE - 2;
  const size_t plane = (size_t)blockIdx.z * (size_t)(GF_H * GF_W);
  const float* gp = g   + plane;
  const float* sp = s   + plane;
  float*       op = (float*)out + plane;

  // ---- Stage A: async-load 36x36 tiles of g and s, zeros outside the image ----
  for (int idx = tid; idx < IN_T * IN_T; idx += NTH) {
    const int i = idx / IN_T, j = idx - i * IN_T;
    const int y = Y0 + i, x = X0 + j;
    if ((unsigned)y < (unsigned)GF_H && (unsigned)x < (unsigned)GF_W) {
      const size_t off = (size_t)y * GF_W + (size_t)x;
      gf_async_g2l(gp + off, &sg[idx]);
      gf_async_g2l(sp + off, &ssr[idx]);
    } else {
      sg[idx]  = 0.0f;   // zero padding (reduce_window pad value)
      ssr[idx] = 0.0f;
    }
  }
  gf_wait_async();      // this wave's async->LDS writes landed
  __syncthreads();      // visible to all waves (also covers the DS zero stores)

  // ---- Stage B: horizontal 3-tap window sums of g, s, g*g, g*s ----
  for (int idx = tid; idx < IN_T * AB_T; idx += NTH) {
    const int i = idx / AB_T, j = idx - i * AB_T;
    const int b0 = i * IN_T + j;
    const float g0 = sg[b0], g1 = sg[b0 + 1], g2 = sg[b0 + 2];
    const float s0 = ssr[b0], s1 = ssr[b0 + 1], s2 = ssr[b0 + 2];
    rg [idx] = g0 + g1 + g2;
    rs [idx] = s0 + s1 + s2;
    rgg[idx] = g0 * g0 + g1 * g1 + g2 * g2;
    rgs[idx] = g0 * s0 + g1 * s1 + g2 * s2;
  }
  __syncthreads();

  // ---- Stage C: vertical sums -> means -> a, b (zero at out-of-image positions) ----
  for (int idx = tid; idx < AB_T * AB_T; idx += NTH) {
    const int i = idx / AB_T, j = idx - i * AB_T;
    const int y = Y0 + 1 + i, x = X0 + 1 + j;
    float a = 0.0f, b = 0.0f;
    if ((unsigned)y < (unsigned)GF_H && (unsigned)x < (unsigned)GF_W) {
      const int r0 = i * AB_T + j;
      const float Sg  = rg [r0] + rg [r0 + AB_T] + rg [r0 + 2 * AB_T];
      const float Ss  = rs [r0] + rs [r0 + AB_T] + rs [r0 + 2 * AB_T];
      const float Sgg = rgg[r0] + rgg[r0 + AB_T] + rgg[r0 + 2 * AB_T];
      const float Sgs = rgs[r0] + rgs[r0 + AB_T] + rgs[r0 + 2 * AB_T];
      const float inv = gf_inv_cnt(y, x);
      const float mx  = Sg  * inv;
      const float my  = Ss  * inv;
      const float mxx = Sgg * inv;
      const float mxy = Sgs * inv;
      const float var = mxx - mx * mx;
      const float cov = mxy - mx * my;
      a = cov / (var + EPS_F);        // IEEE divide: eps matters under cancellation
      b = my - a * mx;
    }
    sa[idx] = a;
    sb[idx] = b;
  }
  __syncthreads();

  // ---- Stage D: horizontal 3-tap sums of a, b ----
  for (int idx = tid; idx < AB_T * TILE; idx += NTH) {
    const int i = idx / TILE, j = idx - i * TILE;
    const int b0 = i * AB_T + j;
    ra[idx] = sa[b0] + sa[b0 + 1] + sa[b0 + 2];
    rb[idx] = sb[b0] + sb[b0 + 1] + sb[b0 + 2];
  }
  __syncthreads();

  // ---- Stage E: vertical sums -> mean_a, mean_b -> out = mean_a*g + mean_b ----
  for (int idx = tid; idx < TILE * TILE; idx += NTH) {
    const int i = idx / TILE, j = idx - i * TILE;
    const int y = Y0 + 2 + i, x = X0 + 2 + j;   // always in-image (1024 % 32 == 0)
    const int r0 = i * TILE + j;
    const float Sa = ra[r0] + ra[r0 + TILE] + ra[r0 + 2 * TILE];
    const float Sb = rb[r0] + rb[r0 + TILE] + rb[r0 + 2 * TILE];
    const float inv = gf_inv_cnt(y, x);
    const float ma = Sa * inv;
    const float mb = Sb * inv;
    const float gv = sg[(i + 2) * IN_T + (j + 2)];
    op[(size_t)y * GF_W + (size_t)x] = ma * gv + mb;
  }
}

extern "C" void kernel_launch(void* const* d_in, const int* in_sizes, int n_in,
                              void* d_out, int out_size, void* d_ws, size_t ws_size,
                              hipStream_t stream) {
  const float* guidance = (const float*)d_in[0];
  const float* src      = (const float*)d_in[1];
  float*       out      = (float*)d_out;
  const int planes = in_sizes[0] / (GF_H * GF_W);   // 8*3 = 24
  dim3 grid(GF_W / TILE, GF_H / TILE, planes);
  dim3 block(NTH);
  guided_filter_fused<<<grid, block, 0, stream>>>(guidance, src, out);
}